// Protos_19292993093657
// MI455X (gfx1250) — compile-verified
//
#include <hip/hip_runtime.h>

#define NUM_CLASSES 21
#define CHANNELS    256
#define HF 128
#define WF 128
#define HL 512
#define WL 512
#define BATCH 16
#define HW (HF * WF)                       // 16384 pixels per plane
#define CHUNK 1024                         // pixels per workgroup
#define CHUNKS_PER_PLANE (HW / CHUNK)      // 16
#define THREADS 256
#define PROTO_ELEMS (NUM_CLASSES * CHANNELS)
#define OUT_ELEMS (PROTO_ELEMS + NUM_CLASSES)

typedef __attribute__((ext_vector_type(2))) float v2f;
typedef __attribute__((ext_vector_type(8))) float v8f;

__global__ void protos_zero_kernel(float* __restrict__ out, int n) {
    int i = blockIdx.x * blockDim.x + threadIdx.x;
    if (i < n) out[i] = 0.0f;
}

// One-hot GEMM segment-sum: sums[k,c] += onehot[k,p] * feat[p,c] via
// V_WMMA_F32_16X16X4_F32 (exact fp32 accumulation, matching the f32 reference).
// Labels are DMA'd into LDS with GLOBAL_LOAD_ASYNC_TO_LDS_B8 (ASYNCcnt path).
__global__ __launch_bounds__(THREADS) void protos_accum_kernel(
    const float* __restrict__ features,       // [B, C, HF, WF] f32
    const long long* __restrict__ labels,     // [B, HL, WL] i64
    float* __restrict__ out) {                // [21*256 sums | 21 counts]
    __shared__ unsigned char lab_s[CHUNK];    // resized labels, one byte each
    __shared__ int hist_s[NUM_CLASSES];

    const int tid = threadIdx.x;
    const int wg  = blockIdx.x;
    const int b          = wg / CHUNKS_PER_PLANE;
    const int chunk_base = (wg % CHUNKS_PER_PLANE) * CHUNK;   // pixel offset in plane

    if (tid < NUM_CLASSES) hist_s[tid] = 0;

    // --- Async byte-gather of nearest-resized labels into LDS ---------------
    // Nearest resize: src row = 4*h, src col = 4*w (512/128 = 4). Classes are
    // 0..20, so the low byte of each little-endian int64 label is the value.
    {
        const unsigned long long gbase =
            (unsigned long long)(labels + (long long)b * (HL * WL));
        #pragma unroll
        for (int j = tid; j < CHUNK; j += THREADS) {
            int p = chunk_base + j;                 // pixel index in 128x128 plane
            int h = p >> 7, w = p & 127;
            unsigned goff = (unsigned)(((h * 4) * WL + (w * 4)) * 8); // byte offset
            unsigned loff = (unsigned)(uintptr_t)&lab_s[j];           // LDS offset
            // GVS mode: copies the byte at SADDR + VADDR to the LDS address in VDST; tracked on ASYNCcnt.
            asm volatile("global_load_async_to_lds_b8 %0, %1, %2"
                         :: "v"(loff), "v"(goff), "s"(gbase) : "memory");
        }
        asm volatile("s_wait_asynccnt 0x0" ::: "memory");
    }
    __syncthreads();

    // Per-WG class histogram from the staged labels (ds_add atomics).
    for (int j = tid; j < CHUNK; j += THREADS)
        atomicAdd(&hist_s[lab_s[j]], 1);
    __syncthreads();

    const int lane = tid & 31;
    const int wave = tid >> 5;
    const int half = lane >> 4;              // K-pair / M-half select per WMMA layout
    const int mrow = lane & 15;              // class row within 16-row A tile
    const int nch  = (lane & 15) + wave * 32;// this lane's channel (first N-tile)

    // Per-lane feature streams: channel nch and nch+16, contiguous pixel run.
    const float* fb0 = features + ((long long)b * CHANNELS + nch) * HW + chunk_base;
    const float* fb1 = fb0 + 16 * HW;
    __builtin_prefetch(fb0, 0, 1);
    __builtin_prefetch(fb1, 0, 1);

    v8f acc00 = {}, acc01 = {}, acc10 = {}, acc11 = {};

    for (int kp = 0; kp < CHUNK; kp += 4) {
        const int ksel = kp + half * 2;      // this lane's K pair (0,1 or 2,3)
        // Two packed labels per lane: aligned 16-bit LDS load.
        unsigned pr = *(const unsigned short*)(lab_s + ksel);
        int l0 = (int)(pr & 0xffu);
        int l1 = (int)(pr >> 8);
        // A tiles (one-hot); rows 21..31 of tile 1 never match (labels < 21).
        v2f a0, a1;
        a0.x = (l0 == mrow)      ? 1.0f : 0.0f;
        a0.y = (l1 == mrow)      ? 1.0f : 0.0f;
        a1.x = (l0 == mrow + 16) ? 1.0f : 0.0f;
        a1.y = (l1 == mrow + 16) ? 1.0f : 0.0f;
        // B tiles: 2 consecutive pixels of this lane's channel (8B load).
        v2f bv0 = *(const v2f*)(fb0 + ksel);
        v2f bv1 = *(const v2f*)(fb1 + ksel);
        acc00 = __builtin_amdgcn_wmma_f32_16x16x4_f32(false, a0, false, bv0,
                                                      (short)0, acc00, false, false);
        acc01 = __builtin_amdgcn_wmma_f32_16x16x4_f32(false, a0, false, bv1,
                                                      (short)0, acc01, false, false);
        acc10 = __builtin_amdgcn_wmma_f32_16x16x4_f32(false, a1, false, bv0,
                                                      (short)0, acc10, false, false);
        acc11 = __builtin_amdgcn_wmma_f32_16x16x4_f32(false, a1, false, bv1,
                                                      (short)0, acc11, false, false);
    }

    // D layout (32-bit 16x16 C/D): VGPR r -> M = r + 8*half, N = lane&15.
    #pragma unroll
    for (int r = 0; r < 8; ++r) {
        int cls0 = r + 8 * half;             // classes 0..15 tile
        atomicAdd(&out[cls0 * CHANNELS + nch],      acc00[r]);
        atomicAdd(&out[cls0 * CHANNELS + nch + 16], acc01[r]);
        int cls1 = cls0 + 16;                // classes 16..31 tile (21..31 padded)
        if (cls1 < NUM_CLASSES) {
            atomicAdd(&out[cls1 * CHANNELS + nch],      acc10[r]);
            atomicAdd(&out[cls1 * CHANNELS + nch + 16], acc11[r]);
        }
    }

    if (tid < NUM_CLASSES)
        atomicAdd(&out[PROTO_ELEMS + tid], (float)hist_s[tid]);
}

__global__ void protos_finalize_kernel(float* __restrict__ out) {
    int i = blockIdx.x * blockDim.x + threadIdx.x;
    if (i < PROTO_ELEMS) {
        float cnt = out[PROTO_ELEMS + i / CHANNELS];
        float v   = out[i];
        out[i] = (cnt > 0.0f) ? (v / cnt) : 0.0f;   // absent classes -> 0
    }
}

extern "C" void kernel_launch(void* const* d_in, const int* in_sizes, int n_in,
                              void* d_out, int out_size, void* d_ws, size_t ws_size,
                              hipStream_t stream) {
    (void)in_sizes; (void)n_in; (void)d_ws; (void)ws_size; (void)out_size;
    const float*     features = (const float*)d_in[0];      // [16,256,128,128] f32
    const long long* labels   = (const long long*)d_in[1];  // [16,512,512] i64
    float*           out      = (float*)d_out;              // 5397 f32

    protos_zero_kernel<<<(OUT_ELEMS + 255) / 256, 256, 0, stream>>>(out, OUT_ELEMS);
    protos_accum_kernel<<<BATCH * CHUNKS_PER_PLANE, THREADS, 0, stream>>>(
        features, labels, out);
    protos_finalize_kernel<<<(PROTO_ELEMS + 255) / 256, 256, 0, stream>>>(out);
}